// Critic_66511863546286
// MI455X (gfx1250) — compile-verified
//
#include <hip/hip_runtime.h>
#include <math.h>

// ---------------------------------------------------------------------------
// Critic attention model, folded form.
//   B=256, S=1024, H=256, static/dynamic channels = 2.
// Precompute: M4[h][4] = [W1@sw | W2@dw], cb[h] = W1@sb + W2@db.
// Per iter:   c[b,h]  = W3 @ ctx[b,:]                (WMMA f32 16x16x4 GEMM)
//             arg     = M4 . x + c + cb              (WMMA f32 16x16x4, K=4!)
//             score   = v . tanh(arg); softmax; ctx = sw @ (attn . static)
// ---------------------------------------------------------------------------

typedef float v2f __attribute__((ext_vector_type(2)));
typedef float v8f __attribute__((ext_vector_type(8)));

#define B_SZ 256
#define S_SZ 1024
#define H_SZ 256

#if defined(__has_builtin)
#if __has_builtin(__builtin_amdgcn_tanhf)
#define TANHF(x) __builtin_amdgcn_tanhf(x)
#endif
#endif
#ifndef TANHF
#define TANHF(x) tanhf(x)
#endif

__device__ __forceinline__ v8f wmma_16x16x4_f32(v2f a, v2f b, v8f c) {
  // 8-arg form: (neg_a, A, neg_b, B, c_mod, C, reuse_a, reuse_b)
  return __builtin_amdgcn_wmma_f32_16x16x4_f32(false, a, false, b, (short)0, c,
                                               false, false);
}

// ---- fold W into rank-2 operators -----------------------------------------
__global__ void prep_kernel(const float* __restrict__ W,
                            const float* __restrict__ sw,
                            const float* __restrict__ sb,
                            const float* __restrict__ dw,
                            const float* __restrict__ db,
                            float* __restrict__ M4, float* __restrict__ cb) {
  const int h = threadIdx.x;
  const float* wr = W + (size_t)h * (3 * H_SZ);
  float a0 = 0.f, a1 = 0.f, a2 = 0.f, a3 = 0.f, bias = 0.f;
  for (int k = 0; k < H_SZ; ++k) {
    float w1 = wr[k];
    float w2 = wr[H_SZ + k];
    a0 += w1 * sw[k * 2];
    a1 += w1 * sw[k * 2 + 1];
    a2 += w2 * dw[k * 2];
    a3 += w2 * dw[k * 2 + 1];
    bias += w1 * sb[k] + w2 * db[k];
  }
  M4[h * 4 + 0] = a0;
  M4[h * 4 + 1] = a1;
  M4[h * 4 + 2] = a2;
  M4[h * 4 + 3] = a3;
  cb[h] = bias;
}

__global__ void ctx_init_kernel(const float* __restrict__ init,
                                const float* __restrict__ sw,
                                const float* __restrict__ sb,
                                float* __restrict__ ctx) {
  const int b = blockIdx.x, h = threadIdx.x;
  ctx[b * H_SZ + h] =
      sw[h * 2] * init[b * 2] + sw[h * 2 + 1] * init[b * 2 + 1] + sb[h];
}

// ---- c = ctx @ W3^T  (256x256x256 f32 GEMM via WMMA 16x16x4) --------------
// One wave per 16x16 output tile. A[m,k]=ctx[m,k], B[k,n]=W[n*768+512+k].
__global__ void ctx_gemm_kernel(const float* __restrict__ ctx,
                                const float* __restrict__ W,
                                float* __restrict__ cmat) {
  const int lane = threadIdx.x & 31;
  const int lo = lane & 15;
  const int hi = lane >> 4;                // 0 or 1 -> K offset 0 / 2
  const int n0 = blockIdx.x * 16;          // h tile
  const int m0 = blockIdx.y * 16;          // batch tile

  v8f acc;
#pragma unroll
  for (int r = 0; r < 8; ++r) acc[r] = 0.0f;

  const float* arow = ctx + (size_t)(m0 + lo) * H_SZ;
  const float* brow = W + (size_t)(n0 + lo) * (3 * H_SZ) + 2 * H_SZ;
#pragma unroll 4
  for (int k0 = 0; k0 < H_SZ; k0 += 4) {
    v2f a, b;
    a.x = arow[k0 + 2 * hi];
    a.y = arow[k0 + 2 * hi + 1];
    b.x = brow[k0 + 2 * hi];
    b.y = brow[k0 + 2 * hi + 1];
    acc = wmma_16x16x4_f32(a, b, acc);
  }
#pragma unroll
  for (int r = 0; r < 8; ++r)
    cmat[(size_t)(m0 + r + 8 * hi) * H_SZ + n0 + lo] = acc[r];
}

// ---- attention iteration: one workgroup (8 waves) per batch ---------------
// Loop order: h-tile outer / s-tile inner. Each wave keeps its 8 s-tile
// B-operands resident in VGPRs and issues 8 independent WMMAs per h-tile.
__global__ void attn_kernel(const float* __restrict__ stat,
                            const float* __restrict__ dyn,
                            const float* __restrict__ sw,
                            const float* __restrict__ sb,
                            const float* __restrict__ vvec,
                            const float* __restrict__ M4,
                            const float* __restrict__ cb,
                            const float* __restrict__ cmat,
                            float* __restrict__ ctx) {
  __shared__ float s_m4[H_SZ * 4];
  __shared__ float s_cc[H_SZ];
  __shared__ float s_vv[H_SZ];
  __shared__ float s_score[S_SZ];
  __shared__ float s_red[16];

  const int b = blockIdx.x;
  const int tid = threadIdx.x;
  const int wave = tid >> 5;
  const int lane = tid & 31;
  const int lo = lane & 15;
  const int hi = lane >> 4;

  s_cc[tid] = cmat[(size_t)b * H_SZ + tid] + cb[tid];
  s_vv[tid] = vvec[tid];
  for (int i = tid; i < H_SZ * 4; i += 256) s_m4[i] = M4[i];
  __syncthreads();

  // B operand source: half-wave hi=0 -> static channels (K=0,1),
  //                   half-wave hi=1 -> dynamic channels (K=2,3).
  const float* xb = hi ? (dyn + (size_t)b * 2 * S_SZ)
                       : (stat + (size_t)b * 2 * S_SZ);

  // Stage this wave's 8 s-tile B operands (s-invariant over h loop).
  v2f bmat[8];
#pragma unroll
  for (int i = 0; i < 8; ++i) {
    const int s0 = (wave + 8 * i) * 16;
    bmat[i].x = xb[s0 + lo];
    bmat[i].y = xb[S_SZ + s0 + lo];
  }

  float part[8];
#pragma unroll
  for (int i = 0; i < 8; ++i) part[i] = 0.f;

  for (int ht = 0; ht < 16; ++ht) {
    const int h0 = ht * 16;
    v2f amat;
    amat.x = s_m4[(h0 + lo) * 4 + 2 * hi];
    amat.y = s_m4[(h0 + lo) * 4 + 2 * hi + 1];
    float cvals[8], vvals[8];
#pragma unroll
    for (int r = 0; r < 8; ++r) {
      cvals[r] = s_cc[h0 + 8 * hi + r];
      vvals[r] = s_vv[h0 + 8 * hi + r];
    }
#pragma unroll
    for (int i = 0; i < 8; ++i) {
      v8f acc;
#pragma unroll
      for (int r = 0; r < 8; ++r) acc[r] = cvals[r];
      acc = wmma_16x16x4_f32(amat, bmat[i], acc);
#pragma unroll
      for (int r = 0; r < 8; ++r) part[i] += vvals[r] * TANHF(acc[r]);
    }
  }

#pragma unroll
  for (int i = 0; i < 8; ++i) {
    part[i] += __shfl_xor(part[i], 16, 32);  // merge M=0..7 with M=8..15
    if (lane < 16) s_score[(wave + 8 * i) * 16 + lo] = part[i];
  }
  __syncthreads();

  // ---- softmax over S ----
  float lmax = -__builtin_inff();
  for (int s = tid; s < S_SZ; s += 256) lmax = fmaxf(lmax, s_score[s]);
  for (int o = 16; o > 0; o >>= 1) lmax = fmaxf(lmax, __shfl_xor(lmax, o, 32));
  if (lane == 0) s_red[wave] = lmax;
  __syncthreads();
  float gmax = s_red[0];
  for (int wv = 1; wv < 8; ++wv) gmax = fmaxf(gmax, s_red[wv]);
  __syncthreads();

  float lsum = 0.f;
  for (int s = tid; s < S_SZ; s += 256) {
    float e = __expf(s_score[s] - gmax);
    s_score[s] = e;
    lsum += e;
  }
  for (int o = 16; o > 0; o >>= 1) lsum += __shfl_xor(lsum, o, 32);
  if (lane == 0) s_red[wave] = lsum;
  __syncthreads();
  float gsum = 0.f;
  for (int wv = 0; wv < 8; ++wv) gsum += s_red[wv];
  const float inv = 1.0f / gsum;
  __syncthreads();

  // ---- ctx = sw @ (attn-weighted static), rank-2 update ----
  float t0 = 0.f, t1 = 0.f;
  const float* sbase = stat + (size_t)b * 2 * S_SZ;
  for (int s = tid; s < S_SZ; s += 256) {
    float a = s_score[s] * inv;
    t0 += a * sbase[s];
    t1 += a * sbase[S_SZ + s];
  }
  for (int o = 16; o > 0; o >>= 1) {
    t0 += __shfl_xor(t0, o, 32);
    t1 += __shfl_xor(t1, o, 32);
  }
  if (lane == 0) { s_red[wave] = t0; s_red[8 + wave] = t1; }
  __syncthreads();
  float T0 = 0.f, T1 = 0.f;
  for (int wv = 0; wv < 8; ++wv) { T0 += s_red[wv]; T1 += s_red[8 + wv]; }

  ctx[(size_t)b * H_SZ + tid] =
      sw[tid * 2] * T0 + sw[tid * 2 + 1] * T1 + sb[tid];
}

// ---- output MLP head ------------------------------------------------------
__global__ void head_kernel(const float* __restrict__ ctx,
                            const float* __restrict__ fc1_w,
                            const float* __restrict__ fc1_b,
                            const float* __restrict__ fc2_w,
                            const float* __restrict__ fc2_b,
                            float* __restrict__ out) {
  const int b = threadIdx.x;  // 256 threads, one per batch row
  const float* row = ctx + (size_t)b * H_SZ;
  float o = fc2_b[0];
  for (int j = 0; j < 20; ++j) {
    float acc = fc1_b[j];
    const float* wrow = fc1_w + (size_t)j * H_SZ;
    for (int k = 0; k < H_SZ; ++k) acc += wrow[k] * row[k];
    o += fc2_w[j] * fmaxf(acc, 0.f);
  }
  out[b] = o;
}

extern "C" void kernel_launch(void* const* d_in, const int* in_sizes, int n_in,
                              void* d_out, int out_size, void* d_ws,
                              size_t ws_size, hipStream_t stream) {
  const float* stat = (const float*)d_in[0];
  const float* dyn  = (const float*)d_in[1];
  const float* init = (const float*)d_in[2];
  const float* sw   = (const float*)d_in[3];
  const float* sb   = (const float*)d_in[4];
  const float* dw   = (const float*)d_in[5];
  const float* db   = (const float*)d_in[6];
  const float* vv   = (const float*)d_in[7];
  const float* W    = (const float*)d_in[8];
  const float* fc1w = (const float*)d_in[9];
  const float* fc1b = (const float*)d_in[10];
  const float* fc2w = (const float*)d_in[11];
  const float* fc2b = (const float*)d_in[12];
  float* out = (float*)d_out;

  // workspace layout (floats): M4[1024] | cb[256] | ctx[65536] | c[65536]
  float* ws  = (float*)d_ws;
  float* M4  = ws;
  float* cb  = ws + 1024;
  float* ctx = ws + 1280;
  float* cm  = ws + 1280 + (size_t)B_SZ * H_SZ;

  prep_kernel<<<1, 256, 0, stream>>>(W, sw, sb, dw, db, M4, cb);
  ctx_init_kernel<<<B_SZ, 256, 0, stream>>>(init, sw, sb, ctx);

  for (int it = 0; it < 3; ++it) {
    ctx_gemm_kernel<<<dim3(H_SZ / 16, B_SZ / 16), 32, 0, stream>>>(ctx, W, cm);
    attn_kernel<<<B_SZ, 256, 0, stream>>>(stat, dyn, sw, sb, vv, M4, cb, cm,
                                          ctx);
  }

  head_kernel<<<1, 256, 0, stream>>>(ctx, fc1w, fc1b, fc2w, fc2b, out);
}